// RBF_17695265259742
// MI455X (gfx1250) — compile-verified
//
#include <hip/hip_runtime.h>
#include <math.h>

// RBF kernel matrix: cov = variance * exp(-0.5 * ||(x_n - xx_m)/scale||^2)
// N = M = 4096, D = 8, all fp32.
//
// ||u-v||^2 = ||u||^2 + ||v||^2 - 2 u.v with u = x/scale, v = xx/scale.
// Cross term is a [4096x8]x[8x4096] fp32 GEMM -> V_WMMA_F32_16X16X4_F32
// (two WMMAs per 16x16 tile cover K=D=8), full fp32 precision.
//
// Roofline: 64 MB fp32 output @ 23.3 TB/s ~ 2.8 us floor (store-bound).
// Epilogue uses native v_exp_f32 (__builtin_amdgcn_exp2f) with all constants
// folded into the base-2 exponent:
//   variance*exp(dot - 0.5(us+vs)) = exp2(dot*log2e + base_row + col_term)
// -> 1 add + 1 fma + 1 trans + 1 store per output element.

#define N_ 4096
#define M_ 4096
#define D_ 8

typedef __attribute__((ext_vector_type(2))) float v2f;
typedef __attribute__((ext_vector_type(8))) float v8f;

__global__ __launch_bounds__(256) void rbf_wmma_kernel(
    const float* __restrict__ x,        // [N, 8]
    const float* __restrict__ xx,       // [M, 8]
    const float* __restrict__ log_scale,// [8]
    const float* __restrict__ log_var,  // [1]
    float* __restrict__ out)            // [N, M]
{
    const float LOG2E  = 1.44269504088896340736f;
    const float NH_L2E = -0.72134752044448170368f;   // -0.5 * log2(e)

    const int lane = threadIdx.x & 31;
    const int wave = threadIdx.x >> 5;          // 0..7
    const int rowTile  = blockIdx.x >> 3;       // 0..255  (16 rows each)
    const int colBlock = blockIdx.x & 7;        // 0..7
    const int colGroup = colBlock * 8 + wave;   // 0..63   (64 cols each)
    const int rowBase  = rowTile * 16;
    const int colBase  = colGroup * 64;

    const int ml   = lane & 15;   // row (A) / col (B) index within tile
    const int half = lane >> 4;   // selects K-half of the fragment

    // Uniform parameters (tiny; cached). Fast native exp is fine here.
    float invs[D_];
#pragma unroll
    for (int d = 0; d < D_; ++d) invs[d] = __expf(-log_scale[d]);
    const float l2var = log_var[0] * LOG2E;      // log2(variance)

    // ---- Load & scale this wave's 16 x-rows (lane L%16 owns row rowBase+L%16).
    const float4* xr = (const float4*)(x + (size_t)(rowBase + ml) * D_);
    float4 xa = xr[0], xb = xr[1];
    float u0 = xa.x * invs[0], u1 = xa.y * invs[1];
    float u2 = xa.z * invs[2], u3 = xa.w * invs[3];
    float u4 = xb.x * invs[4], u5 = xb.y * invs[5];
    float u6 = xb.z * invs[6], u7 = xb.w * invs[7];
    float usq = u0*u0 + u1*u1 + u2*u2 + u3*u3
              + u4*u4 + u5*u5 + u6*u6 + u7*u7;

    // A fragments (16x4 fp32): lane L -> M = L%16, K = 2*(L/16) + {0,1}.
    v2f aLo, aHi;
    aLo.x = half ? u2 : u0;  aLo.y = half ? u3 : u1;   // K = 0..3
    aHi.x = half ? u6 : u4;  aHi.y = half ? u7 : u5;   // K = 4..7

    // Row norms per C layout: VGPR i holds rows i and i+8 (by lane half).
    // Fold -0.5*log2e and log2(variance) in while we're at it.
    float base[8];
#pragma unroll
    for (int i = 0; i < 8; ++i) {
        float rs = __shfl(usq, i + 8 * half, 32);
        base[i] = fmaf(rs, NH_L2E, l2var);
    }

#pragma unroll
    for (int t = 0; t < 4; ++t) {
        const int col = colBase + t * 16 + ml;

        // ---- Load & scale one xx-column per lane.
        const float4* vr = (const float4*)(xx + (size_t)col * D_);
        float4 va = vr[0], vb = vr[1];
        float w0 = va.x * invs[0], w1 = va.y * invs[1];
        float w2 = va.z * invs[2], w3 = va.w * invs[3];
        float w4 = vb.x * invs[4], w5 = vb.y * invs[5];
        float w6 = vb.z * invs[6], w7 = vb.w * invs[7];
        float vsq = w0*w0 + w1*w1 + w2*w2 + w3*w3
                  + w4*w4 + w5*w5 + w6*w6 + w7*w7;

        // B fragments (4x16 fp32): lane L -> N = L%16, K = 2*(L/16) + {0,1}.
        v2f bLo, bHi;
        bLo.x = half ? w2 : w0;  bLo.y = half ? w3 : w1;
        bHi.x = half ? w6 : w4;  bHi.y = half ? w7 : w5;

        // ---- dot = u . v via two fp32 WMMAs (K = 0..3, then K = 4..7).
        v8f acc = {};
        acc = __builtin_amdgcn_wmma_f32_16x16x4_f32(
            false, aLo, false, bLo, (short)0, acc, false, false);
        acc = __builtin_amdgcn_wmma_f32_16x16x4_f32(
            false, aHi, false, bHi, (short)0, acc, false, false);

        // ---- Epilogue: cov = exp2(dot*log2e + base_row + col_term),
        // with variance and both norms pre-folded into the exponent.
        const float colTerm = vsq * NH_L2E;
#pragma unroll
        for (int i = 0; i < 8; ++i) {
            float arg = fmaf(acc[i], LOG2E, base[i] + colTerm);
            float val = __builtin_amdgcn_exp2f(arg);   // native v_exp_f32
            out[(size_t)(rowBase + i + 8 * half) * M_ + col] = val;
        }
    }
}

extern "C" void kernel_launch(void* const* d_in, const int* in_sizes, int n_in,
                              void* d_out, int out_size, void* d_ws, size_t ws_size,
                              hipStream_t stream) {
    (void)in_sizes; (void)n_in; (void)out_size; (void)d_ws; (void)ws_size;
    const float* x   = (const float*)d_in[0];   // [4096, 8]
    const float* xx  = (const float*)d_in[1];   // [4096, 8]
    const float* lsc = (const float*)d_in[2];   // [8]
    const float* lvr = (const float*)d_in[3];   // [1]
    float* out = (float*)d_out;                 // [4096, 4096]

    // 256 row-tiles (16 rows) x 8 col-blocks (512 cols) = 2048 blocks.
    // Each block: 8 waves, each wave one 16x64 strip (8 v_wmma_f32_16x16x4_f32).
    dim3 grid(2048), block(256);
    hipLaunchKernelGGL(rbf_wmma_kernel, grid, block, 0, stream,
                       x, xx, lsc, lvr, out);
}